// HadamardTransform_40664750358773
// MI455X (gfx1250) — compile-verified
//
#include <hip/hip_runtime.h>

// MI455X / gfx1250: wave32, WMMA f32 16x16x4.
// Per batch-row (2304 f32): FWHT-64 on contiguous blocks in registers
// (2 in-reg stages + 4 xor-shuffle stages), then H36 matmul on tensor
// cores: out(48x64 padded) = H36pad(48x36) x t(36x64), scaled by 1/48.

typedef __attribute__((ext_vector_type(2))) float v2f;
typedef __attribute__((ext_vector_type(4))) float v4f;
typedef __attribute__((ext_vector_type(8))) float v8f;

#define WAVES_PER_BLOCK 8
#define ROW_LEN 2304  // 36 * 64

__global__ __launch_bounds__(256)
void fwht_had36_kernel(const float* __restrict__ x,
                       const float* __restrict__ hadk,   // 36x36 +/-1 matrix (row-major, 'kj')
                       float* __restrict__ out,
                       int nrows)
{
    __shared__ __align__(16) float smem[WAVES_PER_BLOCK * ROW_LEN];

    const int lane    = threadIdx.x & 31;
    const int wave    = threadIdx.x >> 5;
    const int halfsel = lane >> 4;    // 0: lanes 0-15, 1: lanes 16-31
    const int lq      = lane & 15;
    const int row     = blockIdx.x * WAVES_PER_BLOCK + wave;   // wave-uniform
    if (row >= nrows) return;         // wave-uniform: EXEC stays all-1s for WMMA

    // ---- A fragments: H36 zero-padded to 48 rows, split into 3 M-tiles x 9 K-steps.
    // ISA 16x4 f32 A layout: VGPR0 = {K0 | K2}, VGPR1 = {K1 | K3}; lane(0-15) = M row.
    v2f afrag[3][9];
    #pragma unroll
    for (int mt = 0; mt < 3; ++mt) {
        const int r  = mt * 16 + lq;
        const int rc = (r < 36) ? r : 35;          // clamp, then select 0 (branchless)
        #pragma unroll
        for (int ks = 0; ks < 9; ++ks) {
            const int k0 = ks * 4 + 2 * halfsel;
            const float hx = hadk[rc * 36 + k0];
            const float hy = hadk[rc * 36 + k0 + 1];
            afrag[mt][ks].x = (r < 36) ? hx : 0.0f;
            afrag[mt][ks].y = (r < 36) ? hy : 0.0f;
        }
    }

    const int wbase = wave * ROW_LEN;              // wave-private LDS slab
    const v4f* xin  = reinterpret_cast<const v4f*>(x + (size_t)row * ROW_LEN);

    // ---- FWHT-64 per contiguous 64-block; each wave does 2 blocks per iteration
    // (block jb = 2*jj + halfsel; element p = 4*lq + e). Streaming (NT) loads.
    #pragma unroll 2
    for (int jj = 0; jj < 18; ++jj) {
        const int jb = jj * 2 + halfsel;
        v4f vv = __builtin_nontemporal_load(&xin[jb * 16 + lq]);

        // in-register stages: p bits 0..1 -> m bits 0..1
        float s0 = vv.x + vv.y, s1 = vv.x - vv.y;
        float s2 = vv.z + vv.w, s3 = vv.z - vv.w;
        float r0 = s0 + s2, r2 = s0 - s2;
        float r1 = s1 + s3, r3 = s1 - s3;

        // cross-lane stages: p bits 2..5 == lane bits 0..3 (masks < 16 keep halves separate)
        #pragma unroll
        for (int st = 1; st <= 8; st <<= 1) {
            const float f = (lane & st) ? -1.0f : 1.0f;   // butterfly sign via FMA
            float p0 = __shfl_xor(r0, st, 32);
            float p1 = __shfl_xor(r1, st, 32);
            float p2 = __shfl_xor(r2, st, 32);
            float p3 = __shfl_xor(r3, st, 32);
            r0 = p0 + f * r0;
            r1 = p1 + f * r1;
            r2 = p2 + f * r2;
            r3 = p3 + f * r3;
        }
        // lane lq, reg e holds t[jb, 4*lq + e] -> contiguous ds_store_b128
        v4f ov; ov.x = r0; ov.y = r1; ov.z = r2; ov.w = r3;
        *reinterpret_cast<v4f*>(&smem[wbase + jb * 64 + 4 * lq]) = ov;
    }
    // Same-wave DS ops are in-order (DScnt); LDS slab is wave-private -> no barrier.

    // ---- WMMA: D(16x16) += A(16x4) x B(4x16), 3 M-tiles x 4 N-tiles x 9 K-steps.
    // B 4x16 layout mirrors A: lane(0-15) = N col; VGPR0 = {K0 | K2}, VGPR1 = {K1 | K3}.
    float* orow = out + (size_t)row * ROW_LEN;
    const float scale = 1.0f / 48.0f;              // 1/sqrt(2304)
    #pragma unroll
    for (int nt = 0; nt < 4; ++nt) {
        v8f c0 = {}, c1 = {}, c2 = {};
        const int mcol = nt * 16 + lq;
        #pragma unroll
        for (int ks = 0; ks < 9; ++ks) {
            const int j0 = ks * 4 + 2 * halfsel;
            v2f b;
            b.x = smem[wbase + j0 * 64 + mcol];
            b.y = smem[wbase + j0 * 64 + 64 + mcol];
            c0 = __builtin_amdgcn_wmma_f32_16x16x4_f32(false, afrag[0][ks], false, b,
                                                       (short)0, c0, false, false);
            c1 = __builtin_amdgcn_wmma_f32_16x16x4_f32(false, afrag[1][ks], false, b,
                                                       (short)0, c1, false, false);
            c2 = __builtin_amdgcn_wmma_f32_16x16x4_f32(false, afrag[2][ks], false, b,
                                                       (short)0, c2, false, false);
        }
        // D layout: VGPR v -> output row M = v + 8*halfsel; lane lq -> col N.
        #pragma unroll
        for (int v = 0; v < 8; ++v) {
            const int kbase = v + 8 * halfsel;
            __builtin_nontemporal_store(c0[v] * scale, &orow[kbase * 64 + mcol]);
            __builtin_nontemporal_store(c1[v] * scale, &orow[(16 + kbase) * 64 + mcol]);
            if (kbase < 4)   // M-tile 2 has only rows 32..35; EXEC restored before next WMMA
                __builtin_nontemporal_store(c2[v] * scale, &orow[(32 + kbase) * 64 + mcol]);
        }
    }
}

extern "C" void kernel_launch(void* const* d_in, const int* in_sizes, int n_in,
                              void* d_out, int out_size, void* d_ws, size_t ws_size,
                              hipStream_t stream) {
    const float* x    = (const float*)d_in[0];
    const float* hadk = (const float*)d_in[1];   // 36*36 floats (+/-1)
    float* out        = (float*)d_out;

    const int nrows  = in_sizes[0] / ROW_LEN;    // 4*8192 = 32768
    const int blocks = (nrows + WAVES_PER_BLOCK - 1) / WAVES_PER_BLOCK;
    hipLaunchKernelGGL(fwht_had36_kernel, dim3(blocks), dim3(256), 0, stream,
                       x, hadk, out, nrows);
}